// GRU_GNN_Model_REF_atten_52467320488321
// MI455X (gfx1250) — compile-verified
//
#include <hip/hip_runtime.h>
#include <cstdint>
#include <cstddef>

#define BATCH 8
#define NV    4096
#define NE    65536
#define SEQ   128
#define IND   480
#define KRFF  240
#define HIDD  960
#define G1C   128
#define GRUC  16

typedef __attribute__((ext_vector_type(2))) float v2f;
typedef __attribute__((ext_vector_type(8))) float v8f;

// ---------------------------------------------------------------------------
// float atomic max via int-max / uint-min (valid IEEE ordering, -inf init)
// ---------------------------------------------------------------------------
__device__ __forceinline__ void atomic_max_f32(float* addr, float val) {
  if (val >= 0.0f) {
    atomicMax((int*)addr, __float_as_int(val));
  } else {
    atomicMin((unsigned int*)addr, __float_as_uint(val));
  }
}

// ---------------------------------------------------------------------------
// fill buffer with -inf (segment-max identity)
// ---------------------------------------------------------------------------
__global__ void k_fill_neginf(float* __restrict__ p, int n) {
  int i = blockIdx.x * blockDim.x + threadIdx.x;
  if (i < n) p[i] = -__builtin_inff();
}

// ---------------------------------------------------------------------------
// in-place: -inf -> 0 (empty segment-max cleanup), keeps WMMA loop clean
// ---------------------------------------------------------------------------
__global__ void k_fixup_neginf(float* __restrict__ p, int n) {
  int i = blockIdx.x * blockDim.x + threadIdx.x;
  if (i < n) {
    float v = p[i];
    if (v == -__builtin_inff()) p[i] = 0.0f;
  }
}

// ---------------------------------------------------------------------------
// Wd = W_top - W_bot for EdgeConv1 (256x128 row-major -> 128x128)
// ---------------------------------------------------------------------------
__global__ void k_wd1(const float* __restrict__ W, float* __restrict__ Wd) {
  int i = blockIdx.x * blockDim.x + threadIdx.x;
  if (i < G1C * G1C) Wd[i] = W[i] - W[i + G1C * G1C];
}

// ---------------------------------------------------------------------------
// EdgeConv0 vertex transform (K=3, tiny; plain VALU)
// P = pos @ (W_top - W_bot),  Q = pos @ W_bot
// ---------------------------------------------------------------------------
__global__ void k_ec0_vertex(const float* __restrict__ pos, const float* __restrict__ W0,
                             float* __restrict__ P, float* __restrict__ Q) {
  int idx = blockIdx.x * blockDim.x + threadIdx.x;       // BATCH*NV*G1C
  if (idx >= BATCH * NV * G1C) return;
  int c = idx & (G1C - 1);
  int v = (idx >> 7) & (NV - 1);
  int b = idx >> 19;
  const float* pp = pos + (size_t)b * (NV * 3) + (size_t)v * 3;
  float p0 = pp[0], p1 = pp[1], p2 = pp[2];
  float wt0 = W0[0 * G1C + c], wt1 = W0[1 * G1C + c], wt2 = W0[2 * G1C + c];
  float wb0 = W0[3 * G1C + c], wb1 = W0[4 * G1C + c], wb2 = W0[5 * G1C + c];
  P[idx] = p0 * (wt0 - wb0) + p1 * (wt1 - wb1) + p2 * (wt2 - wb2);
  Q[idx] = p0 * wb0 + p1 * wb1 + p2 * wb2;
}

// ---------------------------------------------------------------------------
// Per-edge: val = prelu(P[dst] + Q[src] + bias), atomic max at dst
// ---------------------------------------------------------------------------
__global__ void k_edge_agg(const float* __restrict__ P, const float* __restrict__ Q,
                           const float* __restrict__ bias, const float* __restrict__ alpha,
                           const int* __restrict__ srcIdx, const int* __restrict__ dstIdx,
                           float* __restrict__ agg) {
  unsigned int gid = blockIdx.x * blockDim.x + threadIdx.x;  // BATCH*NE*G1C = 2^26
  int c = (int)(gid & (G1C - 1));
  int e = (int)((gid >> 7) & (NE - 1));
  int b = (int)(gid >> 23);
  int s = srcIdx[e];
  int d = dstIdx[e];
  float val = P[((size_t)b * NV + d) * G1C + c] + Q[((size_t)b * NV + s) * G1C + c] + bias[c];
  if (val < 0.0f) val *= alpha[c];
  atomic_max_f32(&agg[((size_t)b * NV + d) * G1C + c], val);
}

// ---------------------------------------------------------------------------
// fp32 WMMA GEMM: C[M,N] = A[M,K] @ B[K,N] (+bias[N]), row-major.
// K, N are compile-time so B-fragment addresses fold into load immediates
// (no per-step 64-bit address arithmetic).
// Block = 128 threads (4 waves) computing a 64x64 C tile:
//   each wave owns 64 rows x 16 cols (4 accumulators, 4 row-subtiles),
//   amortizing the strided B-fragment loads over 4 WMMAs.
// Grid: x = M/64, y = ceil(N/64). Partial 16-col tiles skipped wave-uniformly
// (EXEC stays all-ones for WMMA). M % 64 == 0, K % 4 == 0, N % 16 == 0.
//
// V_WMMA_F32_16X16X4_F32 layouts (ISA 7.12.2):
//   A 16x4:  lane = kh*16 + m (kh=lane>>4), v0=A[m][2kh], v1=A[m][2kh+1]
//   B 4x16:  lane = kh*16 + n,              v0=B[2kh][n], v1=B[2kh+1][n]
//   C/D:     col = lane&15, row(vgpr r) = r + 8*kh
// ---------------------------------------------------------------------------
template <int K, int N>
__global__ void k_gemm_wmma_f32(const float* __restrict__ A, const float* __restrict__ B,
                                float* __restrict__ C, const float* __restrict__ bias) {
  int lane  = threadIdx.x & 31;
  int wave  = threadIdx.x >> 5;
  int ntile = blockIdx.y * 4 + wave;               // 16-col tile index
  if (ntile * 16 >= N) return;                     // wave-uniform
  int kh   = lane >> 4;
  int ml   = lane & 15;
  int row0 = blockIdx.x * 64 + ml;                 // rows row0 + {0,16,32,48}
  int col  = ntile * 16 + ml;

  const float* Ar0 = A + (size_t)(row0 +  0) * K + 2 * kh;
  const float* Ar1 = A + (size_t)(row0 + 16) * K + 2 * kh;
  const float* Ar2 = A + (size_t)(row0 + 32) * K + 2 * kh;
  const float* Ar3 = A + (size_t)(row0 + 48) * K + 2 * kh;
  const float* Bc  = B + (size_t)(2 * kh) * N + col;

  v8f acc0 = {}, acc1 = {}, acc2 = {}, acc3 = {};
#pragma unroll 4
  for (int k = 0; k < K; k += 4) {
    v2f bf;
    bf.x = Bc[(size_t)k * N];
    bf.y = Bc[(size_t)k * N + N];
    v2f a0 = *(const v2f*)(Ar0 + k);
    v2f a1 = *(const v2f*)(Ar1 + k);
    v2f a2 = *(const v2f*)(Ar2 + k);
    v2f a3 = *(const v2f*)(Ar3 + k);
    acc0 = __builtin_amdgcn_wmma_f32_16x16x4_f32(false, a0, false, bf, (short)0, acc0, false, false);
    acc1 = __builtin_amdgcn_wmma_f32_16x16x4_f32(false, a1, false, bf, (short)0, acc1, false, false);
    acc2 = __builtin_amdgcn_wmma_f32_16x16x4_f32(false, a2, false, bf, (short)0, acc2, false, false);
    acc3 = __builtin_amdgcn_wmma_f32_16x16x4_f32(false, a3, false, bf, (short)0, acc3, false, false);
  }

  float bv = bias ? bias[col] : 0.0f;
  float* Cp0 = C + (size_t)(blockIdx.x * 64 +  0 + 8 * kh) * N + col;
  float* Cp1 = C + (size_t)(blockIdx.x * 64 + 16 + 8 * kh) * N + col;
  float* Cp2 = C + (size_t)(blockIdx.x * 64 + 32 + 8 * kh) * N + col;
  float* Cp3 = C + (size_t)(blockIdx.x * 64 + 48 + 8 * kh) * N + col;
#pragma unroll
  for (int r = 0; r < 8; r++) {
    Cp0[(size_t)r * N] = acc0[r] + bv;
    Cp1[(size_t)r * N] = acc1[r] + bv;
    Cp2[(size_t)r * N] = acc2[r] + bv;
    Cp3[(size_t)r * N] = acc3[r] + bv;
  }
}

// ---------------------------------------------------------------------------
// x_in = [cos(2*pi*emb), sin(2*pi*emb), x]
// ---------------------------------------------------------------------------
__global__ void k_build_xin(const float* __restrict__ x, const float* __restrict__ emb,
                            float* __restrict__ xin) {
  int idx = blockIdx.x * blockDim.x + threadIdx.x;  // BATCH*SEQ*HIDD
  if (idx >= BATCH * SEQ * HIDD) return;
  int h  = idx % HIDD;
  int bs = idx / HIDD;
  const float TWO_PI = 6.2831853071795864769f;
  float o;
  if (h < KRFF)        o = cosf(TWO_PI * emb[(size_t)bs * KRFF + h]);
  else if (h < 2*KRFF) o = sinf(TWO_PI * emb[(size_t)bs * KRFF + (h - KRFF)]);
  else                 o = x[(size_t)bs * IND + (h - 2 * KRFF)];
  xin[idx] = o;
}

// ---------------------------------------------------------------------------
// q = queryvec @ Wq + bq  (1x960 @ 960x960, tiny)
// ---------------------------------------------------------------------------
__global__ void k_qproj(const float* __restrict__ qv, const float* __restrict__ Wq,
                        const float* __restrict__ bq, float* __restrict__ q) {
  int j = blockIdx.x * blockDim.x + threadIdx.x;
  if (j >= HIDD) return;
  float acc = bq[j];
  for (int i = 0; i < HIDD; i++) acc += qv[i] * Wq[(size_t)i * HIDD + j];
  q[j] = acc;
}

// ---------------------------------------------------------------------------
// single-query attention per batch: softmax(q.k/sqrt(H)) @ v
// ---------------------------------------------------------------------------
__global__ void k_attention(const float* __restrict__ q, const float* __restrict__ k,
                            const float* __restrict__ v, float* __restrict__ att) {
  __shared__ float sc[SEQ];
  __shared__ float red;
  int b = blockIdx.x;
  int s = threadIdx.x;

  const float* kr = k + ((size_t)b * SEQ + s) * HIDD;
  float dot = 0.0f;
  for (int h = 0; h < HIDD; h++) dot += q[h] * kr[h];
  float score = dot * (1.0f / sqrtf((float)HIDD));
  sc[s] = score;
  __syncthreads();
  if (s == 0) {
    float mx = sc[0];
    for (int i = 1; i < SEQ; i++) mx = fmaxf(mx, sc[i]);
    red = mx;
  }
  __syncthreads();
  float ev = expf(score - red);
  sc[s] = ev;
  __syncthreads();
  if (s == 0) {
    float sm = 0.0f;
    for (int i = 0; i < SEQ; i++) sm += sc[i];
    red = sm;
  }
  __syncthreads();
  float p = ev / red;
  sc[s] = p;
  __syncthreads();

  for (int h = s; h < HIDD; h += SEQ) {
    float acc = 0.0f;
    for (int s2 = 0; s2 < SEQ; s2++)
      acc += sc[s2] * v[((size_t)b * SEQ + s2) * HIDD + h];
    att[(size_t)b * HIDD + h] = acc;
  }
}

// ---------------------------------------------------------------------------
// g = prelu(att @ g1_W + b)   (8x960 @ 960x256)
// ---------------------------------------------------------------------------
__global__ void k_g1(const float* __restrict__ att, const float* __restrict__ W,
                     const float* __restrict__ bias, const float* __restrict__ alpha,
                     float* __restrict__ g) {
  int b = blockIdx.x;
  int j = threadIdx.x;  // 256
  const float* ar = att + (size_t)b * HIDD;
  float acc = bias[j];
  for (int kk = 0; kk < HIDD; kk++) acc += ar[kk] * W[(size_t)kk * 256 + j];
  if (acc < 0.0f) acc *= alpha[j];
  g[b * 256 + j] = acc;
}

// ---------------------------------------------------------------------------
// gru = prelu(g @ g2_W + b)   (8x256 @ 256x65536) -- bandwidth bound;
// W read exactly once, 8 batches accumulated per thread.
// ---------------------------------------------------------------------------
__global__ void k_g2(const float* __restrict__ g, const float* __restrict__ W,
                     const float* __restrict__ bias, const float* __restrict__ alpha,
                     float* __restrict__ gru) {
  __shared__ float gs[BATCH * 256];
  int tid = threadIdx.x;
  for (int i = tid; i < BATCH * 256; i += 256) gs[i] = g[i];
  __syncthreads();
  int j = blockIdx.x * 256 + tid;
  float acc[BATCH];
  float bv = bias[j];
#pragma unroll
  for (int b = 0; b < BATCH; b++) acc[b] = bv;
  for (int kk = 0; kk < 256; kk++) {
    float w = W[(size_t)kk * 65536 + j];
#pragma unroll
    for (int b = 0; b < BATCH; b++) acc[b] += gs[b * 256 + kk] * w;
  }
  float a = alpha[j];
#pragma unroll
  for (int b = 0; b < BATCH; b++) {
    float o = acc[b];
    if (o < 0.0f) o *= a;
    gru[(size_t)b * 65536 + j] = o;
  }
}

// ---------------------------------------------------------------------------
// y = cat[gru(16), h1(128)] @ out_W(144x3) + out_b   (h1 pre-fixed, no -inf)
// ---------------------------------------------------------------------------
__global__ void k_final(const float* __restrict__ gru, const float* __restrict__ h1,
                        const float* __restrict__ Wo, const float* __restrict__ bo,
                        float* __restrict__ out) {
  __shared__ float sW[(GRUC + G1C) * 3];
  int tid = threadIdx.x;
  for (int i = tid; i < (GRUC + G1C) * 3; i += blockDim.x) sW[i] = Wo[i];
  __syncthreads();
  int idx = blockIdx.x * blockDim.x + tid;  // BATCH*NV
  int vtx = idx & (NV - 1);
  int b   = idx >> 12;
  float a0 = bo[0], a1 = bo[1], a2 = bo[2];
  const float* gp = gru + (size_t)b * (NV * GRUC) + (size_t)vtx * GRUC;
#pragma unroll
  for (int t = 0; t < GRUC; t++) {
    float gv = gp[t];
    a0 += gv * sW[t * 3 + 0];
    a1 += gv * sW[t * 3 + 1];
    a2 += gv * sW[t * 3 + 2];
  }
  const float* hp = h1 + ((size_t)b * NV + vtx) * G1C;
  for (int c = 0; c < G1C; c++) {
    float hv = hp[c];
    int r = GRUC + c;
    a0 += hv * sW[r * 3 + 0];
    a1 += hv * sW[r * 3 + 1];
    a2 += hv * sW[r * 3 + 2];
  }
  float* op = out + (size_t)b * (NV * 3) + (size_t)vtx * 3;
  op[0] = a0; op[1] = a1; op[2] = a2;
}

// ---------------------------------------------------------------------------
extern "C" void kernel_launch(void* const* d_in, const int* in_sizes, int n_in,
                              void* d_out, int out_size, void* d_ws, size_t ws_size,
                              hipStream_t stream) {
  (void)in_sizes; (void)n_in; (void)out_size; (void)ws_size;

  const float* x        = (const float*)d_in[0];
  const float* smoothed = (const float*)d_in[1];
  const float* Brff     = (const float*)d_in[2];
  const float* ec0_W    = (const float*)d_in[3];
  const float* ec0_b    = (const float*)d_in[4];
  const float* ec0_a    = (const float*)d_in[5];
  const float* ec1_W    = (const float*)d_in[6];
  const float* ec1_b    = (const float*)d_in[7];
  const float* ec1_a    = (const float*)d_in[8];
  const float* queryvec = (const float*)d_in[9];
  const float* Wq       = (const float*)d_in[10];
  const float* bq       = (const float*)d_in[11];
  const float* Wk       = (const float*)d_in[12];
  const float* bk       = (const float*)d_in[13];
  const float* Wv       = (const float*)d_in[14];
  const float* bv       = (const float*)d_in[15];
  const float* g1_W     = (const float*)d_in[16];
  const float* g1_b     = (const float*)d_in[17];
  const float* g1_a     = (const float*)d_in[18];
  const float* g2_W     = (const float*)d_in[19];
  const float* g2_b     = (const float*)d_in[20];
  const float* g2_a     = (const float*)d_in[21];
  const float* out_W    = (const float*)d_in[22];
  const float* out_b    = (const float*)d_in[23];
  const int*   eidx     = (const int*)d_in[24];   // (2, NE): row0=src, row1=dst
  const int* srcIdx = eidx;
  const int* dstIdx = eidx + NE;

  float* out = (float*)d_out;
  float* ws  = (float*)d_ws;

  // ---- workspace layout (floats); P/Q and agg regions are reused across stages
  const size_t SZ_PQ = (size_t)BATCH * NV * G1C;          // 4,194,304
  float* Pbuf = ws;                                       // P0 -> P1
  float* Qbuf = ws + SZ_PQ;                               // Q0 -> Q1
  float* Agg  = ws + 2 * SZ_PQ;                           // agg0 -> agg1 (h1)
  float* Wd1  = ws + 3 * SZ_PQ;                           // 128*128
  float* Emb  = Wd1 + (size_t)G1C * G1C;                  // 1024*240
  float* Xin  = Emb + (size_t)BATCH * SEQ * KRFF;         // 1024*960
  float* Kb   = Xin + (size_t)BATCH * SEQ * HIDD;         // 1024*960
  float* Vb   = Kb  + (size_t)BATCH * SEQ * HIDD;         // 1024*960
  float* Qv   = Vb  + (size_t)BATCH * SEQ * HIDD;         // 960
  float* Att  = Qv  + HIDD;                               // 8*960
  float* Gb   = Att + (size_t)BATCH * HIDD;               // 8*256
  float* Gru  = Gb  + (size_t)BATCH * 256;                // 8*65536

  // ---- GNN branch ----
  k_wd1<<<(G1C * G1C) / 256, 256, 0, stream>>>(ec1_W, Wd1);
  k_ec0_vertex<<<(BATCH * NV * G1C) / 256, 256, 0, stream>>>(smoothed, ec0_W, Pbuf, Qbuf);
  k_fill_neginf<<<(int)(SZ_PQ / 256), 256, 0, stream>>>(Agg, (int)SZ_PQ);
  k_edge_agg<<<(BATCH * NE * G1C) / 256, 256, 0, stream>>>(Pbuf, Qbuf, ec0_b, ec0_a,
                                                           srcIdx, dstIdx, Agg);
  k_fixup_neginf<<<(int)(SZ_PQ / 256), 256, 0, stream>>>(Agg, (int)SZ_PQ);   // h0 clean
  // EdgeConv1 vertex GEMMs (WMMA): M = B*NV = 32768
  k_gemm_wmma_f32<G1C, G1C><<<dim3((BATCH * NV) / 64, G1C / 64), 128, 0, stream>>>(
      Agg, Wd1, Pbuf, nullptr);
  k_gemm_wmma_f32<G1C, G1C><<<dim3((BATCH * NV) / 64, G1C / 64), 128, 0, stream>>>(
      Agg, ec1_W + (size_t)G1C * G1C, Qbuf, nullptr);
  k_fill_neginf<<<(int)(SZ_PQ / 256), 256, 0, stream>>>(Agg, (int)SZ_PQ);
  k_edge_agg<<<(BATCH * NE * G1C) / 256, 256, 0, stream>>>(Pbuf, Qbuf, ec1_b, ec1_a,
                                                           srcIdx, dstIdx, Agg);
  k_fixup_neginf<<<(int)(SZ_PQ / 256), 256, 0, stream>>>(Agg, (int)SZ_PQ);   // h1 clean

  // ---- attention branch ----
  // emb = x @ B : (1024 x 480) @ (480 x 240)
  k_gemm_wmma_f32<IND, KRFF><<<dim3((BATCH * SEQ) / 64, (KRFF + 63) / 64), 128, 0, stream>>>(
      x, Brff, Emb, nullptr);
  k_build_xin<<<(BATCH * SEQ * HIDD) / 256, 256, 0, stream>>>(x, Emb, Xin);
  // k, v : (1024 x 960) @ (960 x 960) + bias
  k_gemm_wmma_f32<HIDD, HIDD><<<dim3((BATCH * SEQ) / 64, HIDD / 64), 128, 0, stream>>>(
      Xin, Wk, Kb, bk);
  k_gemm_wmma_f32<HIDD, HIDD><<<dim3((BATCH * SEQ) / 64, HIDD / 64), 128, 0, stream>>>(
      Xin, Wv, Vb, bv);
  k_qproj<<<(HIDD + 255) / 256, 256, 0, stream>>>(queryvec, Wq, bq, Qv);
  k_attention<<<BATCH, SEQ, 0, stream>>>(Qv, Kb, Vb, Att);
  k_g1<<<BATCH, 256, 0, stream>>>(Att, g1_W, g1_b, g1_a, Gb);
  k_g2<<<65536 / 256, 256, 0, stream>>>(Gb, g2_W, g2_b, g2_a, Gru);

  // ---- merge ----
  k_final<<<(BATCH * NV) / 128, 128, 0, stream>>>(Gru, Agg, out_W, out_b, out);
}